// SparseEdgeEmbedding_38878043964000
// MI455X (gfx1250) — compile-verified
//
#include <hip/hip_runtime.h>
#include <hip/hip_bf16.h>
#include <math.h>

#define K_NN  32
#define NOUT  16

typedef __attribute__((ext_vector_type(2))) float v2f;
typedef __attribute__((ext_vector_type(8))) float v8f;

// DPP controls (stay within a 16-lane DPP row == our half-wave groups)
#define DPP_QXOR1 0xB1   // quad_perm [1,0,3,2]  -> lane^1
#define DPP_QXOR2 0x4E   // quad_perm [2,3,0,1]  -> lane^2
#define DPP_RHM   0x141  // row_half_mirror      -> lane^7 (completes 8-group)
#define DPP_RM    0x140  // row_mirror           -> lane^15 (completes 16-row)

template <int CTRL>
__device__ __forceinline__ unsigned dpp32(unsigned v) {
    return (unsigned)__builtin_amdgcn_update_dpp(0, (int)v, CTRL, 0xF, 0xF, true);
}
template <int CTRL>
__device__ __forceinline__ unsigned long long dpp_max64(unsigned long long p) {
    unsigned lo = dpp32<CTRL>((unsigned)p);
    unsigned hi = dpp32<CTRL>((unsigned)(p >> 32));
    unsigned long long q = ((unsigned long long)hi << 32) | lo;
    return q > p ? q : p;
}
template <int CTRL>
__device__ __forceinline__ unsigned long long dpp_min64(unsigned long long p) {
    unsigned lo = dpp32<CTRL>((unsigned)p);
    unsigned hi = dpp32<CTRL>((unsigned)(p >> 32));
    unsigned long long q = ((unsigned long long)hi << 32) | lo;
    return q < p ? q : p;
}
template <int CTRL>
__device__ __forceinline__ float dpp_fmax(float v) {
    return fmaxf(v, __uint_as_float(dpp32<CTRL>(__float_as_uint(v))));
}
__device__ __forceinline__ unsigned long long red_max64(unsigned long long p) {
    p = dpp_max64<DPP_QXOR1>(p); p = dpp_max64<DPP_QXOR2>(p);
    p = dpp_max64<DPP_RHM>(p);   p = dpp_max64<DPP_RM>(p);
    return p;
}
__device__ __forceinline__ unsigned long long red_min64(unsigned long long p) {
    p = dpp_min64<DPP_QXOR1>(p); p = dpp_min64<DPP_QXOR2>(p);
    p = dpp_min64<DPP_RHM>(p);   p = dpp_min64<DPP_RM>(p);
    return p;
}
__device__ __forceinline__ float red_fmax(float v) {
    v = dpp_fmax<DPP_QXOR1>(v); v = dpp_fmax<DPP_QXOR2>(v);
    v = dpp_fmax<DPP_RHM>(v);   v = dpp_fmax<DPP_RM>(v);
    return v;
}

// Pack coords + squared norm: c4[j] = (x, y, z, x^2+y^2+z^2)
__global__ void prep_kernel(const float* __restrict__ coord,
                            float4* __restrict__ c4, int n) {
    int j = blockIdx.x * blockDim.x + threadIdx.x;
    if (j < n) {
        float x = coord[3 * j + 0], y = coord[3 * j + 1], z = coord[3 * j + 2];
        c4[j] = make_float4(x, y, z, x * x + y * y + z * z);
    }
}

// Packed (dist_bits | idx | lane | slot): unique per entry; orders by dist,
// then index -> deterministic eviction / extraction tie-breaks.
__device__ __forceinline__ unsigned long long pack_e(float d, int idx, int l15, int slot) {
    return ((unsigned long long)__float_as_uint(d) << 32)
         | ((unsigned long long)((unsigned)idx & 0x3FFFu) << 5)
         | ((unsigned long long)l15 << 1)
         | (unsigned long long)slot;
}

// One 16x16 d^2 tile: WMMA + streaming top-32 update.
__device__ __forceinline__ void process_tile(
    int j0, float4 cb, int half, int l15, unsigned long long hm,
    v2f A, const float (&srow)[8],
    float (&e0)[8], float (&e1)[8], float (&maxv)[8],
    int (&id0)[8], int (&id1)[8])
{
    // B-matrix (4x16 f32): lanes 0-15 hold (K0,K1)=(x,y), lanes 16-31
    // hold (K2,K3)=(z,0) of column j0+l15.
    v2f B;
    B.x = half ? cb.z : cb.x;
    B.y = half ? 0.0f : cb.y;
    float sc = cb.w;

    v8f acc;
#pragma unroll
    for (int r = 0; r < 8; ++r) acc[r] = srow[r] + sc;

    // d2 tile = |xi|^2 + |xj|^2 - 2*xi.xj
    acc = __builtin_amdgcn_wmma_f32_16x16x4_f32(false, A, false, B,
                                                (short)0, acc, false, false);

#pragma unroll
    for (int r = 0; r < 8; ++r) {
        float d2 = acc[r];
        d2 = d2 < 0.0f ? 0.0f : d2;
        unsigned long long bal = __ballot(d2 < maxv[r]) & hm;  // half-uniform
        while (bal) {
            int src = __ffsll((long long)bal) - 1;
            bal &= bal - 1;
            float cval = __shfl(d2, src);   // divergent-index broadcast
            int   cidx = j0 + (src & 15);
            if (cval < maxv[r]) {
                // evict packed-max entry (largest dist, then largest idx)
                unsigned long long p0 = pack_e(e0[r], id0[r], l15, 0);
                unsigned long long p1 = pack_e(e1[r], id1[r], l15, 1);
                unsigned long long pm = red_max64(p0 > p1 ? p0 : p1);
                if ((int)((pm >> 1) & 15ull) == l15) {
                    if ((pm & 1ull) == 0) { e0[r] = cval; id0[r] = cidx; }
                    else                  { e1[r] = cval; id1[r] = cidx; }
                }
                // refresh running 32nd-smallest (pure-VALU DPP reduce)
                maxv[r] = red_fmax(fmaxf(e0[r], e1[r]));
            }
        }
    }
}

__launch_bounds__(128)
__global__ void knn_wmma_kernel(const float4* __restrict__ c4,
                                const float* __restrict__ sigma_range,
                                int* __restrict__ out_idx,
                                float* __restrict__ out_val,
                                int n) {
    const int lane = threadIdx.x & 31;
    const int l15  = lane & 15;
    const int half = lane >> 4;
    const int wv   = (threadIdx.x >> 5) + blockIdx.x * (blockDim.x >> 5);
    const int i0   = wv * 16;
    if (i0 >= n) return;

    float sig    = sigma_range[l15];
    float inv2s2 = 1.0f / (2.0f * sig * sig);

    // A-matrix (16x4 f32): lanes 0-15 hold (K0,K1)=(x,y), lanes 16-31 hold
    // (K2,K3)=(z,0) of row i0+l15. Scaled by -2 so WMMA yields -2*dot.
    float4 ra = c4[i0 + l15];
    v2f A;
    A.x = half ? (-2.0f * ra.z) : (-2.0f * ra.x);
    A.y = half ? 0.0f           : (-2.0f * ra.y);

    // per-register row squared norms (uniform within each half)
    float srow[8];
#pragma unroll
    for (int r = 0; r < 8; ++r)
        srow[r] = c4[i0 + r + half * 8].w;

    // top-32 state per row: 2 slots/lane across a 16-lane half
    float e0[8], e1[8], maxv[8];
    int   id0[8], id1[8];
#pragma unroll
    for (int r = 0; r < 8; ++r) {
        e0[r] = INFINITY; e1[r] = INFINITY; maxv[r] = INFINITY;
        id0[r] = -1;      id1[r] = -1;
    }

    const unsigned long long hm = half ? 0xFFFF0000ull : 0x0000FFFFull;

    // Software-pipelined column sweep: keep the next pair of 16-col tiles in
    // flight while processing the current pair (low occupancy -> the wave
    // itself must hide L2 latency).
    float4 cb0 = c4[l15];
    float4 cb1 = c4[16 + l15];
    for (int j0 = 0; j0 < n; j0 += 32) {
        int jn = (j0 + 32 < n) ? (j0 + 32) : 0;   // wrap: last prefetch unused
        float4 nb0 = c4[jn + l15];
        float4 nb1 = c4[jn + 16 + l15];

        process_tile(j0,      cb0, half, l15, hm, A, srow, e0, e1, maxv, id0, id1);
        process_tile(j0 + 16, cb1, half, l15, hm, A, srow, e0, e1, maxv, id0, id1);

        cb0 = nb0; cb1 = nb1;
    }

    // extract ascending (dist, idx) and emit indices + 16-sigma weights
    const int NK = n * K_NN;
#pragma unroll
    for (int r = 0; r < 8; ++r) {
        int row = i0 + r + half * 8;
#pragma unroll 1
        for (int t = 0; t < K_NN; ++t) {
            unsigned long long p0 = pack_e(e0[r], id0[r], l15, 0);
            unsigned long long p1 = pack_e(e1[r], id1[r], l15, 1);
            unsigned long long pm = red_min64(p0 < p1 ? p0 : p1);
            if ((int)((pm >> 1) & 15ull) == l15) {       // owner removes entry
                if ((pm & 1ull) == 0) { e0[r] = INFINITY; id0[r] = -1; }
                else                  { e1[r] = INFINITY; id1[r] = -1; }
            }
            float d2v = __uint_as_float((unsigned)(pm >> 32));
            int   idx = (int)((pm >> 5) & 0x3FFFull);
            float kd  = sqrtf(d2v);                      // reference: sqrt then square
            float w   = expf(-(kd * kd) * inv2s2);
            if (!(w > 0.1f)) w = 0.0f;                   // also kills NaN
            int e = row * K_NN + t;
            out_val[(long long)e * NOUT + l15] = w;
            if (l15 == 0) {
                out_idx[e]          = 0;    // batch
                out_idx[NK + e]     = row;  // row
                out_idx[2 * NK + e] = idx;  // col
            }
        }
    }
}

extern "C" void kernel_launch(void* const* d_in, const int* in_sizes, int n_in,
                              void* d_out, int out_size, void* d_ws, size_t ws_size,
                              hipStream_t stream) {
    const float* coord = (const float*)d_in[0];
    const float* sigma = (const float*)d_in[1];
    int n = in_sizes[0] / 3;                 // 16384

    float4* c4      = (float4*)d_ws;         // n * 16 B = 256 KB scratch
    int*    out_idx = (int*)d_out;           // [3, n*32] int32
    float*  out_val = (float*)d_out + (size_t)3 * n * K_NN;  // [n*32, 16] f32

    prep_kernel<<<(n + 255) / 256, 256, 0, stream>>>(coord, c4, n);

    int waves  = n / 16;                     // 1024 row tiles
    int blocks = (waves + 3) / 4;            // 4 waves / 128-thread block
    knn_wmma_kernel<<<blocks, 128, 0, stream>>>(c4, sigma, out_idx, out_val, n);
}